// QauntMBBlock_13142599926137
// MI455X (gfx1250) — compile-verified
//
#include <hip/hip_runtime.h>
#include <hip/hip_bf16.h>

// ---------------------------------------------------------------------------
// QauntMBBlock for MI455X (gfx1250, wave32, WMMA).
// Shapes fixed by the reference: B=8, C=256, H=W=64 (HW=4096), TOTAL=768.
// Heavy GEMMs (qkv 1x1 conv, proj 1x1 conv) run on V_WMMA_F32_16X16X4_F32
// with double-buffered async-to-LDS staging of the B operand (ASYNCcnt).
// ---------------------------------------------------------------------------

typedef __attribute__((ext_vector_type(2))) float v2f;
typedef __attribute__((ext_vector_type(8))) float v8f;
typedef __attribute__((ext_vector_type(2))) int   v2i_t;

#define HW    4096
#define CIN   256
#define TOT   768
#define BATCH 8

// gfx1250 async global->LDS path (ASYNCcnt), with safe fallback.
#if defined(__has_builtin)
#  if __has_builtin(__builtin_amdgcn_global_load_async_to_lds_b64) && \
      __has_builtin(__builtin_amdgcn_s_wait_asynccnt)
#    define USE_ASYNC_LDS 1
#  endif
#endif
#ifndef USE_ASYNC_LDS
#  define USE_ASYNC_LDS 0
#endif

__device__ __forceinline__ void stage_b64(const float* gsrc, float* lds_dst) {
#if USE_ASYNC_LDS
    __builtin_amdgcn_global_load_async_to_lds_b64(
        (__attribute__((address_space(1))) v2i_t*)gsrc,
        (__attribute__((address_space(3))) v2i_t*)lds_dst,
        0, 0);
#else
    *reinterpret_cast<float2*>(lds_dst) = *reinterpret_cast<const float2*>(gsrc);
#endif
}

// ---------------------------------------------------------------------------
// GEMM: O[z][m][n] = sum_k A[m*lda + k] * Bm[z*strideB + k*4096 + n]
//       (+ Res[z*strideR + m*4096 + n] if Res != nullptr)
//
// Block: 256 threads = 8 waves arranged 4(M) x 2(N) super-tiles of 32x32;
// block tile 128(M) x 64(N). Each wave holds a 2x2 grid of 16x16 f32
// accumulators -> 4 WMMAs per K-substep with A/B fragment reuse.
// B tile (8K x 64N) staged in LDS, DOUBLE-BUFFERED: the async copy of tile
// i+1 is issued before computing tile i, and s_wait_asynccnt(1) only waits
// for the older copy (ASYNCcnt completes in order) -> copy/compute overlap.
//
// f32 WMMA lane layout (ISA 7.12.2):
//   A (16x4): lane l<16 -> M=l, VGPR0=K0,VGPR1=K1; lane l>=16 -> M=l-16, K2/K3
//   B (4x16): lane l<16 -> N=l, rows K0/K1 in v0/v1; lanes>=16 -> rows K2/K3
//   C (16x16): VGPR i -> M = i + 8*(lane>=16), N = lane&15
// ---------------------------------------------------------------------------
__global__ __launch_bounds__(256)
void wmma_gemm_kernel(const float* __restrict__ A,
                      const float* __restrict__ Bm,
                      const float* __restrict__ Res,
                      float* __restrict__ O,
                      int Kdim, int lda,
                      size_t strideB, size_t strideO, size_t strideR) {
    const int lane = threadIdx.x & 31;
    const int wave = threadIdx.x >> 5;
    const int half = lane >> 4;          // 0: K{0,1}/M 0..7, 1: K{2,3}/M 8..15
    const int l    = lane & 15;
    const int tr   = wave >> 1;          // 0..3  (M super-tile)
    const int tc   = wave & 1;           // 0..1  (N super-tile)

    const int m0   = blockIdx.y * 128 + tr * 32;   // wave's first output row
    const int tcol = tc * 32;                      // wave's first col in block

    const float* Bp = Bm + (size_t)blockIdx.z * strideB + (size_t)blockIdx.x * 64;
    float*       Op = O  + (size_t)blockIdx.z * strideO;

    __shared__ float Bs[2][8 * 64];                // double-buffered 8K x 64N

    // Stager mapping: thread t fills Bs[buf][kk][nl..nl+1]
    const int skk = threadIdx.x >> 5;              // 0..7
    const int snl = (threadIdx.x & 31) * 2;        // 0..62 (even)
    const float* gB = Bp + snl + (size_t)skk * HW;
    const int sofs = skk * 64 + snl;

    v8f c[2][2];
    #pragma unroll
    for (int mi = 0; mi < 2; ++mi)
        #pragma unroll
        for (int nj = 0; nj < 2; ++nj)
            c[mi][nj] = (v8f){0.f, 0.f, 0.f, 0.f, 0.f, 0.f, 0.f, 0.f};

    // Prologue: start copy of tile 0 into buffer 0.
    stage_b64(gB, &Bs[0][sofs]);

    int buf = 0;
    for (int kb = 0; kb < Kdim; kb += 8) {
        // Issue the async copy of the NEXT tile into the alternate buffer,
        // then wait only for the OLDER outstanding copy (current tile).
        if (kb + 8 < Kdim) {
            const float* nsrc = gB + (size_t)(kb + 8) * HW;
            stage_b64(nsrc, &Bs[buf ^ 1][sofs]);
            __builtin_prefetch(nsrc + (size_t)8 * HW, 0, 3);
#if USE_ASYNC_LDS
            __builtin_amdgcn_s_wait_asynccnt(1);
#endif
        } else {
#if USE_ASYNC_LDS
            __builtin_amdgcn_s_wait_asynccnt(0);
#endif
        }
        __syncthreads();

        // ---- 2 K-substeps of 4, 4 WMMAs each ----
        const float* Bcur = Bs[buf];
        #pragma unroll
        for (int ks = 0; ks < 2; ++ks) {
            const int kk = kb + ks * 4;

            v2f a[2];
            #pragma unroll
            for (int mi = 0; mi < 2; ++mi) {
                const float2 av = *reinterpret_cast<const float2*>(
                    A + (size_t)(m0 + mi * 16 + l) * lda + kk + 2 * half);
                a[mi].x = av.x; a[mi].y = av.y;
            }

            v2f bf[2];
            #pragma unroll
            for (int nj = 0; nj < 2; ++nj) {
                const int col = tcol + nj * 16 + l;
                const int row = ks * 4 + 2 * half;
                bf[nj].x = Bcur[row * 64 + col];        // ds_load_b32
                bf[nj].y = Bcur[(row + 1) * 64 + col];  // ds_load_b32
            }

            #pragma unroll
            for (int mi = 0; mi < 2; ++mi)
                #pragma unroll
                for (int nj = 0; nj < 2; ++nj)
                    c[mi][nj] = __builtin_amdgcn_wmma_f32_16x16x4_f32(
                        false, a[mi], false, bf[nj],
                        (short)0, c[mi][nj], false, false);
        }
        // Tile consumed: safe for iteration i+1 to overwrite this buffer's
        // partner (written above) only after all waves pass this barrier.
        __syncthreads();
        buf ^= 1;
    }

    // ---- epilogue ----
    #pragma unroll
    for (int mi = 0; mi < 2; ++mi) {
        #pragma unroll
        for (int nj = 0; nj < 2; ++nj) {
            #pragma unroll
            for (int i = 0; i < 8; ++i) {
                const int   m  = m0 + mi * 16 + half * 8 + i;
                const int   n  = blockIdx.x * 64 + tcol + nj * 16 + l;
                const size_t ix = (size_t)m * HW + n;
                float v = c[mi][nj][i];
                if (Res) v += Res[(size_t)blockIdx.z * strideR + ix];
                Op[ix] = v;
            }
        }
    }
}

// ---------------------------------------------------------------------------
// Fused depthwise 5x5 (groups=768, pad 2) + grouped 1x1 (96 groups, 8->8).
// One thread = one pixel of one group; avoids storing the dw intermediate.
// ---------------------------------------------------------------------------
__global__ __launch_bounds__(256)
void agg_fused_kernel(const float* __restrict__ qkv,
                      const float* __restrict__ Wdw,
                      const float* __restrict__ Wpw,
                      float* __restrict__ agg) {
    const int px = blockIdx.x * blockDim.x + threadIdx.x;  // 0..4095
    const int g  = blockIdx.y;                             // 0..95
    const int b  = blockIdx.z;                             // 0..7
    const int y  = px >> 6, x = px & 63;

    const float* src = qkv + ((size_t)b * TOT + (size_t)g * 8) * HW;

    float dwv[8];
    #pragma unroll
    for (int i = 0; i < 8; ++i) {
        const float* w = Wdw + (size_t)(g * 8 + i) * 25;   // uniform -> s_loads
        const float* p = src + (size_t)i * HW;
        float acc = 0.f;
        #pragma unroll
        for (int dy = 0; dy < 5; ++dy) {
            const int yy = y + dy - 2;
            if (yy < 0 || yy > 63) continue;
            #pragma unroll
            for (int dx = 0; dx < 5; ++dx) {
                const int xx = x + dx - 2;
                if (xx < 0 || xx > 63) continue;
                acc = fmaf(p[yy * 64 + xx], w[dy * 5 + dx], acc);
            }
        }
        dwv[i] = acc;
    }

    float* dst = agg + ((size_t)b * TOT + (size_t)g * 8) * HW + px;
    #pragma unroll
    for (int o = 0; o < 8; ++o) {
        const float* w = Wpw + (size_t)(g * 8 + o) * 8;    // uniform -> s_loads
        float acc = 0.f;
        #pragma unroll
        for (int i = 0; i < 8; ++i) acc = fmaf(w[i], dwv[i], acc);
        dst[(size_t)o * HW] = acc;
    }
}

// ---------------------------------------------------------------------------
// kv[b,h,d,e] = sum_n relu(K[n,d]) * [V|1][n,e]   (d<8, e<9, n<4096)
// Head h<32 reads qkv channels h*24+..., h>=32 reads agg channels (h-32)*24+...
// Per-thread 72 accumulators -> wave32 shuffle tree -> LDS ds_add_f32 combine.
// ---------------------------------------------------------------------------
__global__ __launch_bounds__(256)
void kv_reduce_kernel(const float* __restrict__ qkv,
                      const float* __restrict__ agg,
                      float* __restrict__ kvbuf) {
    const int h = blockIdx.x;   // 0..63
    const int b = blockIdx.y;   // 0..7

    const float* base = (h < 32)
        ? qkv + ((size_t)b * TOT + (size_t)h * 24) * HW
        : agg + ((size_t)b * TOT + (size_t)(h - 32) * 24) * HW;
    const float* kptr = base + (size_t)8  * HW;
    const float* vptr = base + (size_t)16 * HW;

    float acc[72];
    #pragma unroll
    for (int t = 0; t < 72; ++t) acc[t] = 0.f;

    for (int n = threadIdx.x; n < HW; n += 256) {
        float kk[8], vv[9];
        #pragma unroll
        for (int d = 0; d < 8; ++d) kk[d] = fmaxf(kptr[(size_t)d * HW + n], 0.f);
        #pragma unroll
        for (int e = 0; e < 8; ++e) vv[e] = vptr[(size_t)e * HW + n];
        vv[8] = 1.f;
        #pragma unroll
        for (int d = 0; d < 8; ++d)
            #pragma unroll
            for (int e = 0; e < 9; ++e)
                acc[d * 9 + e] = fmaf(kk[d], vv[e], acc[d * 9 + e]);
    }

    // wave32 tree reduction
    #pragma unroll
    for (int t = 0; t < 72; ++t) {
        float v = acc[t];
        v += __shfl_down(v, 16, 32);
        v += __shfl_down(v, 8, 32);
        v += __shfl_down(v, 4, 32);
        v += __shfl_down(v, 2, 32);
        v += __shfl_down(v, 1, 32);
        acc[t] = v;
    }

    __shared__ float sm[72];
    for (int t = threadIdx.x; t < 72; t += 256) sm[t] = 0.f;
    __syncthreads();
    if ((threadIdx.x & 31) == 0) {
        #pragma unroll
        for (int t = 0; t < 72; ++t) atomicAdd(&sm[t], acc[t]);   // ds_add_f32
    }
    __syncthreads();

    float* out = kvbuf + ((size_t)b * 64 + h) * 72;
    for (int t = threadIdx.x; t < 72; t += 256) out[t] = sm[t];
}

// ---------------------------------------------------------------------------
// out[n,e] = (sum_d relu(Q[n,d]) * kv[d,e]) / (sum_d relu(Q[n,d])*kv[d,8]+eps)
// kv is uniform per block -> scalar loads. Output layout (B, 64, 8, 4096).
// ---------------------------------------------------------------------------
__global__ __launch_bounds__(256)
void attn_apply_kernel(const float* __restrict__ qkv,
                       const float* __restrict__ agg,
                       const float* __restrict__ kvbuf,
                       float* __restrict__ attn) {
    const int px = blockIdx.x * blockDim.x + threadIdx.x;
    const int h  = blockIdx.y;   // 0..63
    const int b  = blockIdx.z;   // 0..7

    const float* base = (h < 32)
        ? qkv + ((size_t)b * TOT + (size_t)h * 24) * HW
        : agg + ((size_t)b * TOT + (size_t)(h - 32) * 24) * HW;
    const float* kv = kvbuf + ((size_t)b * 64 + h) * 72;   // uniform -> s_loads

    float q[8];
    #pragma unroll
    for (int d = 0; d < 8; ++d) q[d] = fmaxf(base[(size_t)d * HW + px], 0.f);

    float denom = 0.f;
    #pragma unroll
    for (int d = 0; d < 8; ++d) denom = fmaf(q[d], kv[d * 9 + 8], denom);
    const float inv = 1.f / (denom + 1e-15f);

    float* dst = attn + (((size_t)b * 64 + h) * 8) * HW + px;
    #pragma unroll
    for (int e = 0; e < 8; ++e) {
        float s = 0.f;
        #pragma unroll
        for (int d = 0; d < 8; ++d) s = fmaf(q[d], kv[d * 9 + e], s);
        dst[(size_t)e * HW] = s * inv;
    }
}

// ---------------------------------------------------------------------------
// Launch. Inputs: x, W_qkv, W_agg_dw, W_agg_pw, W_proj (all fp32).
// Workspace layout (floats): qkv[8*768*4096] agg[8*768*4096]
//                            kv[8*64*72] attn[8*512*4096]  (~269 MB total)
// ---------------------------------------------------------------------------
extern "C" void kernel_launch(void* const* d_in, const int* in_sizes, int n_in,
                              void* d_out, int out_size, void* d_ws, size_t ws_size,
                              hipStream_t stream) {
    const float* x     = (const float*)d_in[0];
    const float* Wqkv  = (const float*)d_in[1];
    const float* Wdw   = (const float*)d_in[2];
    const float* Wpw   = (const float*)d_in[3];
    const float* Wproj = (const float*)d_in[4];
    float* out = (float*)d_out;

    float* ws   = (float*)d_ws;
    float* qkv  = ws;                                   // 25,165,824 floats
    float* agg  = qkv + (size_t)BATCH * TOT * HW;       // 25,165,824
    float* kvb  = agg + (size_t)BATCH * TOT * HW;       //     36,864
    float* attn = kvb + (size_t)BATCH * 64 * 72;        // 16,777,216

    dim3 blk(256);

    // 1) qkv = W_qkv (768x256) * x (256x4096) per batch
    wmma_gemm_kernel<<<dim3(HW / 64, TOT / 128, BATCH), blk, 0, stream>>>(
        Wqkv, x, nullptr, qkv,
        /*Kdim=*/CIN, /*lda=*/CIN,
        (size_t)CIN * HW, (size_t)TOT * HW, 0);

    // 2) fused dw5x5 + grouped pw
    agg_fused_kernel<<<dim3(HW / 256, 96, BATCH), blk, 0, stream>>>(
        qkv, Wdw, Wpw, agg);

    // 3) kv = relu(K)^T [V|1]
    kv_reduce_kernel<<<dim3(64, BATCH), blk, 0, stream>>>(qkv, agg, kvb);

    // 4) normalized linear attention apply
    attn_apply_kernel<<<dim3(HW / 256, 64, BATCH), blk, 0, stream>>>(
        qkv, agg, kvb, attn);

    // 5) proj (256 x 512 used columns of W_proj, lda=1536) + residual x
    wmma_gemm_kernel<<<dim3(HW / 64, CIN / 128, BATCH), blk, 0, stream>>>(
        Wproj, attn, x, out,
        /*Kdim=*/512, /*lda=*/1536,
        (size_t)512 * HW, (size_t)CIN * HW, (size_t)CIN * HW);
}